// StyleTreeGenerator_10617159156365
// MI455X (gfx1250) — compile-verified
//
#include <hip/hip_runtime.h>

// ---------------------------------------------------------------------------
// StyleTreeGenerator for MI455X (gfx1250), fp32 via V_WMMA_F32_16X16X4_F32.
// ---------------------------------------------------------------------------

typedef float v2f __attribute__((ext_vector_type(2)));
typedef float v8f __attribute__((ext_vector_type(8)));

#define GF_LEAKY   1
#define GF_ACCUM   2
#define GF_BIASCOL 4

__device__ __forceinline__ float leakyf(float x) { return x > 0.0f ? x : 0.2f * x; }

__device__ __forceinline__ void store_elem(float* __restrict__ C, long idx, float v,
                                           float b, int flags) {
    v += b;
    if (flags & GF_ACCUM) v += C[idx];
    if (flags & GF_LEAKY) v = leakyf(v);
    C[idx] = v;
}

// Generic GEMM: C[r,n] (+)= act( sum_k A[arow(r),k] * B[k,n] + bias[n] )
// - one wave per 32x32 C tile (2x2 of 16x16 WMMA tiles), 4 waves per block.
// - blockIdx.z = "node" with per-matrix node strides (batched einsums).
// - rep>1: ancestor row remap arow = b*inNodes + (node_out / rep)  (jnp.repeat).
// - R must be a multiple of 32 (always true here, B=32); K multiple of 4.
// - FULL=true: N is a multiple of 32 -> no column masking anywhere (hot path).
template <bool FULL>
__global__ __launch_bounds__(128)
void wmma_gemm_f32(const float* __restrict__ A, const float* __restrict__ B,
                   float* __restrict__ C, const float* __restrict__ bias,
                   int R, int K, int N, int lda, int ldb, int ldc,
                   long aNS, long bNS, long cNS,
                   int outNodes, int rep, int flags)
{
    const int lane = threadIdx.x;       // 0..31 (wave32)
    const int wave = threadIdx.y;       // 0..3
    const long node = blockIdx.z;
    A += node * aNS; B += node * bNS; C += node * cNS;

    const int m0 = blockIdx.x * 32;
    const int n0 = (blockIdx.y * 4 + wave) * 32;
    if (n0 >= N) return;                // wave-uniform early out (EXEC stays full)

    const int ml = lane & 15;
    const int kh = (lane >> 4) << 1;    // A/B lane K sub-offset: 0 or 2

    // A fragment rows (16x4 layout: lanes 0-15 M=0..15 K={0,1}; lanes 16-31 K={2,3})
    int r0 = m0 + ml, r1 = r0 + 16;
    int ar0 = r0, ar1 = r1;
    if (rep > 1) {
        int inN = outNodes / rep;
        int b0 = r0 / outNodes; ar0 = b0 * inN + (r0 - b0 * outNodes) / rep;
        int b1 = r1 / outNodes; ar1 = b1 * inN + (r1 - b1 * outNodes) / rep;
    }
    const float* a0p = A + (long)ar0 * lda + kh;
    const float* a1p = A + (long)ar1 * lda + kh;

    // B fragment cols (4x16 layout: lanes 0-15 N=col K={0,1}; lanes 16-31 K={2,3})
    const int nc0 = n0 + ml, nc1 = nc0 + 16;
    const bool ok0 = FULL || (nc0 < N), ok1 = FULL || (nc1 < N);
    const float msk0 = ok0 ? 1.f : 0.f, msk1 = ok1 ? 1.f : 0.f;
    const float* b0p = B + (ok0 ? nc0 : 0) + (long)kh * ldb;
    const float* b1p = B + (ok1 ? nc1 : 0) + (long)kh * ldb;

    v8f acc00 = {}, acc01 = {}, acc10 = {}, acc11 = {};
    for (int k = 0; k < K; k += 4) {
        long ko = (long)k * ldb;
        v2f a0, a1, bv0, bv1;
        a0.x = a0p[k];  a0.y = a0p[k + 1];
        a1.x = a1p[k];  a1.y = a1p[k + 1];
        if (FULL) {
            bv0.x = b0p[ko];  bv0.y = b0p[ko + ldb];
            bv1.x = b1p[ko];  bv1.y = b1p[ko + ldb];
        } else {
            bv0.x = b0p[ko] * msk0;  bv0.y = b0p[ko + ldb] * msk0;
            bv1.x = b1p[ko] * msk1;  bv1.y = b1p[ko + ldb] * msk1;
        }
        acc00 = __builtin_amdgcn_wmma_f32_16x16x4_f32(false, a0, false, bv0, (short)0, acc00, false, false);
        acc01 = __builtin_amdgcn_wmma_f32_16x16x4_f32(false, a0, false, bv1, (short)0, acc01, false, false);
        acc10 = __builtin_amdgcn_wmma_f32_16x16x4_f32(false, a1, false, bv0, (short)0, acc10, false, false);
        acc11 = __builtin_amdgcn_wmma_f32_16x16x4_f32(false, a1, false, bv1, (short)0, acc11, false, false);
    }

    // C layout: VGPR v -> row m0 + v + (lane>=16 ? 8 : 0), col = lane&15
    const int rs = (lane >> 4) << 3;
    float bb0 = 0.f, bb1 = 0.f;
    if (flags & GF_BIASCOL) {
        if (ok0) bb0 = bias[nc0];
        if (ok1) bb1 = bias[nc1];
    }
#pragma unroll
    for (int v = 0; v < 8; ++v) {
        long rA = m0 + rs + v;
        long rB = rA + 16;
        if (ok0) { store_elem(C, rA * ldc + nc0, acc00[v], bb0, flags);
                   store_elem(C, rB * ldc + nc0, acc10[v], bb0, flags); }
        if (ok1) { store_elem(C, rA * ldc + nc1, acc01[v], bb1, flags);
                   store_elem(C, rB * ldc + nc1, acc11[v], bb1, flags); }
    }
}

// y[i] = leaky(y[i] + bias[i % C])   (fused depth-6 map epilogue)
__global__ void bias_act_kernel(float* __restrict__ y, const float* __restrict__ bias,
                                int Cc, int total) {
    int i = blockIdx.x * blockDim.x + threadIdx.x;
    if (i >= total) return;
    y[i] = leakyf(y[i] + bias[i % Cc]);
}

// out[b,j,c] = act( ysup[b,j,c] + yanc[b, j/deg, c] + bias2[j%deg, c] )
__global__ void combine_kernel(const float* __restrict__ ysup, const float* __restrict__ yanc,
                               const float* __restrict__ bias2, float* __restrict__ out,
                               int J, int Cc, int deg, int total, int act) {
    int i = blockIdx.x * blockDim.x + threadIdx.x;
    if (i >= total) return;
    int c = i % Cc; int t = i / Cc; int j = t % J; int b = t / J;
    float v = ysup[i] + yanc[((long)b * (J / deg) + j / deg) * Cc + c];
    if (act) v = leakyf(v + bias2[(j % deg) * Cc + c]);
    out[i] = v;
}

// Per (b,c): mean/var over node axis (population var, like jnp.var)
__global__ void stats_kernel(const float* __restrict__ X, float* __restrict__ mean,
                             float* __restrict__ var, int Bn, int Nn, int Cc) {
    int t = blockIdx.x * blockDim.x + threadIdx.x;
    if (t >= Bn * Cc) return;
    int b = t / Cc, c = t % Cc;
    const float* p = X + (long)b * Nn * Cc + c;
    float s = 0.f, s2 = 0.f;
    for (int j = 0; j < Nn; ++j) { float v = p[(long)j * Cc]; s += v; s2 += v * v; }
    float m = s / (float)Nn;
    mean[t] = m;
    var[t] = fmaxf(s2 / (float)Nn - m * m, 0.f);
}

// X = (X - mx)/sqrt(vx+eps) * sqrt(vs+eps) + ms   (in place)
__global__ void adain_kernel(float* __restrict__ X,
                             const float* __restrict__ mx, const float* __restrict__ vx,
                             const float* __restrict__ ms, const float* __restrict__ vs,
                             int J, int Cc, int total) {
    int i = blockIdx.x * blockDim.x + threadIdx.x;
    if (i >= total) return;
    int c = i % Cc; int b = i / (J * Cc);
    int s = b * Cc + c;
    float sc = sqrtf(vs[s] + 1e-5f) * rsqrtf(vx[s] + 1e-5f);
    X[i] = (X[i] - mx[s]) * sc + ms[s];
}

// ---------------------------------------------------------------------------
extern "C" void kernel_launch(void* const* d_in, const int* in_sizes, int n_in,
                              void* d_out, int out_size, void* d_ws, size_t ws_size,
                              hipStream_t stream)
{
    static const int F[8]      = {96, 256, 256, 256, 128, 128, 128, 3};
    static const int DEGc[7]   = {1, 2, 2, 2, 2, 2, 64};
    static const int LSc[7]    = {4, 2, 1, 1, 1, 1, 1};
    static const int NODESc[7] = {1, 1, 2, 4, 8, 16, 32};
    static const int tnodes[8] = {1, 1, 2, 4, 8, 16, 32, 2048};
    const int Bb = 32;

    // ---- unpack inputs (setup_inputs() insertion order, depth-first) ----
    int idx = 0;
    auto nxt = [&]() { return (const float*)d_in[idx++]; };
    const float* zin0 = nxt();          // z  (32,1,96)
    const float* x0   = nxt();          // x0 (32,1,96)

    const float* mapBranch[7] = {nullptr,nullptr,nullptr,nullptr,nullptr,nullptr,nullptr};
    const float* mapW[7][4];  const float* mapBias[7][4];
    for (int d = 0; d < 7; ++d) {
        if (DEGc[d] > 1) mapBranch[d] = nxt();
        for (int l = 0; l < LSc[d]; ++l) mapW[d][l] = nxt();
        for (int l = 0; l < LSc[d]; ++l) mapBias[d][l] = nxt();
    }
    const float* ancW[7][7];
    const float* synBranch[7]; const float* supW1[7]; const float* supW2[7]; const float* synBias[7];
    for (int d = 0; d < 7; ++d) {
        for (int i = 0; i <= d; ++i) ancW[d][i] = nxt();
        synBranch[d] = nxt(); supW1[d] = nxt(); supW2[d] = nxt(); synBias[d] = nxt();
    }

    // ---- workspace carve-up (fp32) ----
    const bool big = ws_size >= (size_t)48 * 1024 * 1024;  // 46.4MB needed for big path
    float* ws = (float*)d_ws;
    size_t off = 0;
    auto carve = [&](size_t n) { float* p = ws + off; off += (n + 63) & ~(size_t)63; return p; };
    float* zA   = carve(262144);
    float* zB   = carve(262144);
    float* xb   = carve(big ? 8388608 : 262144);   // depth-6 branch einsum output
    float* Hb   = carve(1310720);                  // support hidden (d<6)
    float* BW   = carve(786432);                   // fused depth-6 map weight (32,8192,3)
    float* Ysup = carve(196608);
    float* Yanc = carve(65536);
    float* W12  = carve(384);                      // sup_W1 @ sup_W2 at d=6 (128x3)
    float* mxb  = carve(8192);  float* vxb = carve(8192);
    float* msb  = carve(8192);  float* vsb = carve(8192);
    float* tbuf[8];
    tbuf[0] = (float*)x0;
    for (int i = 1; i <= 6; ++i) tbuf[i] = carve((size_t)Bb * tnodes[i] * F[i]);
    tbuf[7] = (float*)d_out;

    auto gemm = [&](const float* A, const float* Bm, float* C, const float* bias,
                    int R, int K, int N, int lda, int ldb, int ldc,
                    long aNS, long bNS, long cNS, int nodes,
                    int outNodes, int rep, int flags) {
        dim3 grid(R / 32, (N + 127) / 128, nodes), blk(32, 4);
        if ((N & 31) == 0)
            wmma_gemm_f32<true><<<grid, blk, 0, stream>>>(A, Bm, C, bias, R, K, N,
                lda, ldb, ldc, aNS, bNS, cNS, outNodes, rep, flags);
        else
            wmma_gemm_f32<false><<<grid, blk, 0, stream>>>(A, Bm, C, bias, R, K, N,
                lda, ldb, ldc, aNS, bNS, cNS, outNodes, rep, flags);
    };

    const float* zcur = zin0;
    float* zping = zA; float* zpong = zB;
    int nz = 1;

    for (int d = 0; d < 7; ++d) {
        const int fin = F[d], fout = F[d + 1], deg = DEGc[d], nod = NODESc[d];

        // ================= mapping block =================
        if (d < 6) {
            if (deg > 1) {  // per-node branching einsum: (32,fin) @ (fin, fin*deg)
                int g = fin * deg;
                gemm(zcur, mapBranch[d], zping, nullptr,
                     Bb, fin, g, nz * fin, g, nz * g,
                     fin, (long)fin * g, g, nz, 0, 1, 0);
                zcur = zping; { float* t = zping; zping = zpong; zpong = t; }
                nz *= deg;
            }
            for (int l = 0; l < LSc[d]; ++l) {  // dense + bias + leaky
                int nn = (l == LSc[d] - 1) ? fout : fin;
                gemm(zcur, mapW[d][l], zping, mapBias[d][l],
                     Bb * nz, fin, nn, fin, nn, nn, 0, 0, 0, 1, 0, 1,
                     GF_BIASCOL | GF_LEAKY);
                zcur = zping; { float* t = zping; zping = zpong; zpong = t; }
            }
        } else {
            // Fuse branching (32,128,8192) with final dense (128,3):
            // BW[n,(k,j),c] = sum_f branching[n,k,j*128+f] * W[f,c]  -> per node (8192,128)@(128,3)
            gemm(mapBranch[6], mapW[6][0], BW, nullptr,
                 8192, 128, 3, 128, 3, 3,
                 (long)128 * 8192, 0, (long)8192 * 3, 32, 0, 1, 0);
            // zout[b,node,(j,c)] = z[b,node,:] @ BW[node]  (128 x 192 per node)
            gemm(zcur, BW, zping, nullptr,
                 Bb, 128, 192, nz * 128, 192, nz * 192,
                 128, (long)8192 * 3, 192, nz, 0, 1, 0);
            int total = Bb * 2048 * 3;
            bias_act_kernel<<<(total + 255) / 256, 256, 0, stream>>>(zping, mapBias[6][0], 3, total);
            zcur = zping; { float* t = zping; zping = zpong; zpong = t; }
            nz *= deg;  // 2048
        }

        // ================= tree-GCN block =================
        // 1) ancestor sum with repeat (fused into GEMM row remap; accumulate)
        for (int i = 0; i <= d; ++i) {
            int rep = nod / tnodes[i];
            gemm(tbuf[i], ancW[d][i], Yanc, nullptr,
                 Bb * nod, F[i], fout, F[i], fout, fout,
                 0, 0, 0, 1, nod, rep, (i == 0 ? 0 : GF_ACCUM));
        }
        // 2) per-node branch einsum + leaky ; 3) support MLP
        int rows = Bb * nod * deg;
        if (d < 6) {
            int g = fin * deg, h = fin * 10;
            gemm(tbuf[d], synBranch[d], xb, nullptr,
                 Bb, fin, g, nod * fin, g, nod * g,
                 fin, (long)fin * g, g, nod, 0, 1, GF_LEAKY);
            gemm(xb, supW1[d], Hb, nullptr, rows, fin, h, fin, h, h, 0, 0, 0, 1, 0, 1, 0);
            gemm(Hb, supW2[d], Ysup, nullptr, rows, h, fout, h, fout, fout, 0, 0, 0, 1, 0, 1, 0);
        } else {
            // W12 = sup_W1 @ sup_W2  (128x1280 @ 1280x3), then Ysup = leaky-branch @ W12
            gemm(supW1[6], supW2[6], W12, nullptr, 128, 1280, 3, 1280, 3, 3, 0, 0, 0, 1, 0, 1, 0);
            if (big) {
                gemm(tbuf[6], synBranch[6], xb, nullptr,
                     Bb, 128, 8192, 32 * 128, 8192, (long)32 * 8192,
                     128, (long)128 * 8192, 8192, 32, 0, 1, GF_LEAKY);
                gemm(xb, W12, Ysup, nullptr, rows, 128, 3, 128, 3, 3, 0, 0, 0, 1, 0, 1, 0);
            } else {
                for (int nd = 0; nd < 32; ++nd) {  // chunked: 1MB scratch per node
                    gemm(tbuf[6] + nd * 128, synBranch[6] + (long)nd * 128 * 8192, xb, nullptr,
                         Bb, 128, 8192, 32 * 128, 8192, 8192, 0, 0, 0, 1, 0, 1, GF_LEAKY);
                    // xb is (b,8192); per batch b: 64 rows (j) of 128, out row b*2048+nd*64+j
                    gemm(xb, W12, Ysup + (long)nd * 192, nullptr,
                         64, 128, 3, 128, 3, 3, 8192, 0, 6144, 32, 0, 1, 0);
                }
            }
        }
        // 4) combine: Ysup + repeat(Yanc,deg) (+ tiled bias, leaky if d<6)
        int J = nod * deg, total = Bb * J * fout;
        combine_kernel<<<(total + 255) / 256, 256, 0, stream>>>(
            Ysup, Yanc, synBias[d], tbuf[d + 1], J, fout, deg, total, d < 6 ? 1 : 0);
        // 5) AdaIN with style = mapped z
        int st = Bb * fout;
        stats_kernel<<<(st + 255) / 256, 256, 0, stream>>>(tbuf[d + 1], mxb, vxb, Bb, J, fout);
        stats_kernel<<<(st + 255) / 256, 256, 0, stream>>>(zcur, msb, vsb, Bb, nz, fout);
        adain_kernel<<<(total + 255) / 256, 256, 0, stream>>>(
            tbuf[d + 1], mxb, vxb, msb, vsb, J, fout, total);
    }

    (void)in_sizes; (void)n_in; (void)out_size;
}